// WaterNet0313_38268158608161
// MI455X (gfx1250) — compile-verified
//
#include <hip/hip_runtime.h>
#include <hip/hip_bf16.h>
#include <stdint.h>

#define NT    1095
#define NS    4096
#define NH    16
#define WARM  365
#define CHUNK 73                 // 1095 = 15 * 73
#define NCHUNK 15
#define ROWF  96                 // 16 sites * 6 channels, contiguous floats per time-row per block
#define BUFF  (CHUNK * ROWF)     // 7008 floats = 28032 B per buffer

typedef __attribute__((ext_vector_type(4))) unsigned int v4u;
typedef __attribute__((ext_vector_type(8))) int          v8i;
typedef __attribute__((ext_vector_type(4))) int          v4i;

// Byte offset of a __shared__ pointer within the group segment (AS(3) offset).
__device__ __forceinline__ unsigned lds_byte_off(const void* p) {
    return (unsigned)(unsigned long long)(uintptr_t)
        (const __attribute__((address_space(3))) char*)p;
}

// 16-lane (half-wave) reductions via ds_swizzle XOR patterns (stay within lanes 0-15 / 16-31).
__device__ __forceinline__ float hsum16(float v) {
    v += __int_as_float(__builtin_amdgcn_ds_swizzle(__float_as_int(v), 0x041f)); // SWAPX1
    v += __int_as_float(__builtin_amdgcn_ds_swizzle(__float_as_int(v), 0x081f)); // SWAPX2
    v += __int_as_float(__builtin_amdgcn_ds_swizzle(__float_as_int(v), 0x101f)); // SWAPX4
    v += __int_as_float(__builtin_amdgcn_ds_swizzle(__float_as_int(v), 0x201f)); // SWAPX8
    return v;
}
__device__ __forceinline__ float hmax16(float v) {
    v = fmaxf(v, __int_as_float(__builtin_amdgcn_ds_swizzle(__float_as_int(v), 0x041f)));
    v = fmaxf(v, __int_as_float(__builtin_amdgcn_ds_swizzle(__float_as_int(v), 0x081f)));
    v = fmaxf(v, __int_as_float(__builtin_amdgcn_ds_swizzle(__float_as_int(v), 0x101f)));
    v = fmaxf(v, __int_as_float(__builtin_amdgcn_ds_swizzle(__float_as_int(v), 0x201f)));
    return v;
}

__device__ __forceinline__ float sigmoidf(float x) {
    return 1.0f / (1.0f + expf(-x));
}

// Issue a TDM 2D-tile DMA: x[t0 .. t0+CHUNK-1, s0 .. s0+15, 0..5] -> LDS buffer.
// Tile row = 96 contiguous f32 (384 B), row stride = NS*6 elements.
__device__ __forceinline__ void tdm_load_chunk(const float* x, int s0, int t0,
                                               unsigned ldsOff) {
    uint64_t ga = (uint64_t)(uintptr_t)x
                + ((uint64_t)t0 * (uint64_t)(NS * 6) + (uint64_t)s0 * 6ull) * 4ull;
    v4u g0;
    g0[0] = 1u;                                              // count=1, user mode
    g0[1] = ldsOff;                                          // lds_addr (bytes)
    g0[2] = (unsigned)ga;                                    // global_addr[31:0]
    g0[3] = (unsigned)((ga >> 32) & 0x1FFFFFFull) | (2u << 30); // addr[56:32] | type=2
    v8i g1;
    g1[0] = (int)(2u << 16);                // data_size = 4 B; no multicast/pad/iterate
    g1[1] = (int)((unsigned)ROWF << 16);    // tensor_dim0[15:0] = 96
    g1[2] = (int)((unsigned)NT   << 16);    // tensor_dim0[31:16]=0 | tensor_dim1 = 1095
    g1[3] = (int)((unsigned)ROWF << 16);    // tensor_dim1[31:16]=0 | tile_dim0 = 96
    g1[4] = CHUNK;                          // tile_dim1 = 73, tile_dim2 = 0
    g1[5] = NS * 6;                         // tensor_dim0_stride = 24576 elements
    g1[6] = 0;                              // stride0[47:32] | stride1[15:0]
    g1[7] = 0;                              // stride1[47:16]
    v4i z4 = {0, 0, 0, 0};                  // groups 2/3 unused (2D tensor)
    v8i z8 = {0, 0, 0, 0, 0, 0, 0, 0};      // trailing descriptor group (unused)
    __builtin_amdgcn_tensor_load_to_lds(g0, g1, z4, z4, z8, 0);
}

__global__ __launch_bounds__(256)
void waternet_scan_kernel(const float* __restrict__ x,
                          const float* __restrict__ w_raw,
                          const float* __restrict__ fm_raw,
                          const float* __restrict__ fi_raw,
                          const float* __restrict__ kp_raw,
                          const float* __restrict__ ks_raw,
                          const float* __restrict__ kg_raw,
                          const float* __restrict__ gr_raw,
                          const float* __restrict__ gL_raw,
                          const float* __restrict__ qb_raw,
                          float* __restrict__ out) {
    __shared__ float tile[2 * BUFF];   // 56064 B double buffer

    const int tid  = threadIdx.x;
    const int sl   = tid >> 4;                 // local site 0..15
    const int h0   = tid & 15;                 // hidden index (lane within 16-group)
    const int s0   = blockIdx.x * 16;
    const int s    = s0 + sl;
    const int idx  = blockIdx.x * 256 + tid;   // == s*NH + h0, fully coalesced
    const int wave = tid >> 5;

    // ---- parameter activations (once, held in VGPRs for the whole scan) ----
    const float fm = expf(fm_raw[idx]);
    const float fi = sigmoidf(fi_raw[idx]);
    const float kp = sigmoidf(kp_raw[idx]);
    const float ks = sigmoidf(ks_raw[idx]);
    const float kg = sigmoidf(kg_raw[idx]) * 0.1f;
    const float gr = sigmoidf(gr_raw[idx]);
    const float gL = expf(gL_raw[idx]) * 2.0f;
    const float qb = fmaxf(qb_raw[idx], 0.0f) * 0.1f;
    const float omgr = 1.0f - gr;

    // softmax over the 16 hidden units of this site (16-lane swizzle reduction)
    const float wr = w_raw[idx];
    const float wmax = hmax16(wr);
    const float we   = expf(wr - wmax);
    const float w    = we / hsum16(we);

    float Sf = 0.0f, Ss = 0.0f, Sd = 0.0f;

    const unsigned lds0 = lds_byte_off(&tile[0]);

    // ---- TDM pipeline: wave 0 drives the DMA, everyone computes ----
    if (wave == 0) {
        tdm_load_chunk(x, s0, 0, lds0);
    }

    for (int c = 0; c < NCHUNK; ++c) {
        if (wave == 0) {
            __builtin_amdgcn_s_wait_tensorcnt(0);       // chunk c landed in LDS
        }
        __syncthreads();                                // publish + retire buffer c-1
        if (wave == 0 && (c + 1) < NCHUNK) {            // prefetch chunk c+1
            tdm_load_chunk(x, s0, (c + 1) * CHUNK,
                           lds0 + (unsigned)(((c + 1) & 1) * BUFF * 4));
        }

        const float* buf = &tile[(c & 1) * BUFF];
        const int tbase = c * CHUNK;

        #pragma unroll 1
        for (int r = 0; r < CHUNK; ++r) {
            // x[t, s, 0..3]; 16 lanes of a site broadcast-read the same LDS words
            const float2* xp = (const float2*)(buf + r * ROWF + sl * 6);
            const float2 c01 = xp[0];          // Prcp_raw, Evp_raw
            const float2 c23 = xp[1];          // T1_raw,  T2_raw

            const float Prcp = fmaxf(c01.x, 0.0f);
            const float ev   = fmaxf(c01.y, 0.0f) * 0.2f;
            const float tm   = 5.0f * (c23.x + c23.y);   // 0.5*(10*T1 + 10*T2)
            const float vf   = sigmoidf(tm);
            const float ps   = Prcp * (1.0f - vf);
            const float pl   = Prcp * vf;

            // bucket recurrence
            const float qf = fminf(Sf + ps, fm * fmaxf(tm, 0.0f));
            Sf = Sf + ps - qf;
            const float S1 = fmaxf(Ss + pl * fi + qf - ev, 0.0f);
            const float qp = fmaxf(S1 - gL, 0.0f) * kp;
            const float qs = fminf(S1, gL) * ks;
            Ss = S1 - qp - qs;
            const float S2 = Sd + qs * gr;
            const float qd = fminf(S2 * kg + qb, S2);
            Sd = S2 - qd;

            const int t = tbase + r;
            if (t >= WARM) {
                float q = hsum16(w * (qp + qs * omgr + qd));
                if (h0 == 0) {
                    out[(size_t)(t - WARM) * NS + s] = q;
                }
            }
        }
    }
}

extern "C" void kernel_launch(void* const* d_in, const int* in_sizes, int n_in,
                              void* d_out, int out_size, void* d_ws, size_t ws_size,
                              hipStream_t stream) {
    (void)in_sizes; (void)n_in; (void)d_ws; (void)ws_size; (void)out_size;
    const float* xx = (const float*)d_in[0];
    const float* w_raw  = (const float*)d_in[1];
    const float* fm_raw = (const float*)d_in[2];
    const float* fi_raw = (const float*)d_in[3];
    const float* kp_raw = (const float*)d_in[4];
    const float* ks_raw = (const float*)d_in[5];
    const float* kg_raw = (const float*)d_in[6];
    const float* gr_raw = (const float*)d_in[7];
    const float* gL_raw = (const float*)d_in[8];
    const float* qb_raw = (const float*)d_in[9];
    float* out = (float*)d_out;

    dim3 grid(NS / 16);   // 256 blocks, 16 sites each
    dim3 block(256);      // 8 waves; lane = hidden unit, 2 sites per wave
    waternet_scan_kernel<<<grid, block, 0, stream>>>(
        xx, w_raw, fm_raw, fi_raw, kp_raw, ks_raw, kg_raw, gr_raw, gL_raw, qb_raw, out);
}